// HGTLayer_85263690760917
// MI455X (gfx1250) — compile-verified
//
#include <hip/hip_runtime.h>
#include <math.h>

// ---------------------------------------------------------------------------
// HGT layer for gfx1250 (MI455X). Dense matmuls via V_WMMA_F32_16X16X4_F32;
// GEMM tiles staged into LDS with GLOBAL_LOAD_ASYNC_TO_LDS_B128 (ASYNCcnt).
// Pipeline: init -> kqv GEMM -> edge attention (WMMA + shfl reduce + atomicMax)
//           -> softmax normalize -> edge aggregate (WMMA + atomicAdd scatter)
//           -> mean finalize -> output GEMM + gated skip.
// ---------------------------------------------------------------------------

namespace {

constexpr int    Nn   = 100000;
constexpr int    IND  = 128;
constexpr int    OUTD = 128;
constexpr int    HH   = 8;
constexpr int    DKK  = 16;
constexpr int    RRr  = 4;
constexpr int    EEe  = 131072;
constexpr float  INV_SQRT_DK = 0.25f;   // 1/sqrt(16)

typedef float v2f __attribute__((ext_vector_type(2)));
typedef float v8f __attribute__((ext_vector_type(8)));

__device__ inline v8f wmma4(v2f a, v2f b, v8f c) {
  // D = A(16x4, f32) * B(4x16, f32) + C(16x16, f32)
  return __builtin_amdgcn_wmma_f32_16x16x4_f32(
      /*neg_a=*/false, a, /*neg_b=*/false, b,
      /*c_mod=*/(short)0, c, /*reuse_a=*/false, /*reuse_b=*/false);
}

__device__ inline v8f zero8() {
  v8f c;
#pragma unroll
  for (int i = 0; i < 8; ++i) c[i] = 0.0f;
  return c;
}

// Async copy 16B global -> LDS (GVS mode: sgpr base + 32-bit vgpr offset).
__device__ inline void async_ld16(unsigned lds_off, unsigned long long base,
                                  unsigned goff) {
  asm volatile("global_load_async_to_lds_b128 %0, %1, %2"
               :
               : "v"(lds_off), "v"(goff), "s"(base)
               : "memory");
}
__device__ inline void wait_async0() {
  asm volatile("s_wait_asynccnt 0" ::: "memory");
}

// Order-preserving float->uint encoding so atomicMax(uint) == float max.
__device__ inline unsigned encF(float f) {
  unsigned u = __float_as_uint(f);
  return (u & 0x80000000u) ? ~u : (u | 0x80000000u);
}
__device__ inline float decF(unsigned u) {
  unsigned v = (u & 0x80000000u) ? (u & 0x7fffffffu) : ~u;
  return __uint_as_float(v);
}

// ---------------------------------------------------------------------------
__global__ void init_zero_kernel(unsigned* __restrict__ p, size_t n) {
  size_t i = (size_t)blockIdx.x * blockDim.x + threadIdx.x;
  const size_t stride = (size_t)gridDim.x * blockDim.x;
  for (; i < n; i += stride) p[i] = 0u;
}

// ---------------------------------------------------------------------------
// Shared 64-row x 128-col GEMM body: 8 waves, wave w owns col tile [16w,16w+16),
// 4 row tiles of 16 accumulated per wave.  K swept in 4 panels of 32 staged in
// LDS via async b128 loads; every B fetch is reused by 4 WMMAs.
// acc[rt][v] maps to row (rt*16 + v + 8*hi), col (n0 + m).
template <typename STORE>
__device__ inline void gemm64(const float* __restrict__ A,  // [>=N,128]
                              const float* __restrict__ W,  // [128,128]
                              int row0, int t, STORE&& store) {
  __shared__ float xs[64 * 32];    // x panel  (64 rows x 32 k)
  __shared__ float wsp[32 * 128];  // W panel  (32 k  x 128 cols)
  const int l  = t & 31;
  const int m  = l & 15;
  const int hi = l >> 4;
  const int n0 = (t >> 5) * 16;

  const unsigned xs_base = (unsigned)(size_t)&xs[0];
  const unsigned ws_base = (unsigned)(size_t)&wsp[0];
  const unsigned long long ab = (unsigned long long)(size_t)A;
  const unsigned long long wb = (unsigned long long)(size_t)W;

  v8f acc[4];
#pragma unroll
  for (int rt = 0; rt < 4; ++rt) acc[rt] = zero8();

  for (int kp = 0; kp < 128; kp += 32) {
    // stage x panel: 512 x 16B chunks (2 per thread), rows clamped to N-1
#pragma unroll
    for (int j = 0; j < 2; ++j) {
      const int c   = t * 2 + j;
      int       gr  = row0 + (c >> 3);
      if (gr > Nn - 1) gr = Nn - 1;
      const unsigned goff = (unsigned)gr * 512u + (unsigned)kp * 4u +
                            (unsigned)(c & 7) * 16u;
      async_ld16(xs_base + (unsigned)c * 16u, ab, goff);
    }
    // stage W panel: 1024 x 16B chunks (4 per thread), contiguous
#pragma unroll
    for (int j = 0; j < 4; ++j) {
      const int c = t * 4 + j;
      async_ld16(ws_base + (unsigned)c * 16u,
                 wb, (unsigned)kp * 512u + (unsigned)c * 16u);
    }
    wait_async0();
    __syncthreads();

#pragma unroll
    for (int k0 = 0; k0 < 32; k0 += 4) {
      const int kk = k0 + 2 * hi;
      v2f b;
      b[0] = wsp[kk * 128 + n0 + m];
      b[1] = wsp[(kk + 1) * 128 + n0 + m];
#pragma unroll
      for (int rt = 0; rt < 4; ++rt) {
        v2f a;
        a[0] = xs[(m + 16 * rt) * 32 + kk];
        a[1] = xs[(m + 16 * rt) * 32 + kk + 1];
        acc[rt] = wmma4(a, b, acc[rt]);
      }
    }
    __syncthreads();
  }

#pragma unroll
  for (int rt = 0; rt < 4; ++rt)
#pragma unroll
    for (int v = 0; v < 8; ++v) {
      const int row = row0 + rt * 16 + v + 8 * hi;
      if (row < Nn) store(row, n0 + m, acc[rt][v]);
    }
}

// K/Q/V projection: out = x @ W + b.  grid=(ceil(N/64), 3), block=256.
__global__ __launch_bounds__(256) void kqv_kernel(
    const float* __restrict__ x,
    const float* __restrict__ Wk, const float* __restrict__ bk,
    const float* __restrict__ Wq, const float* __restrict__ bq,
    const float* __restrict__ Wv, const float* __restrict__ bv,
    float* __restrict__ Kb, float* __restrict__ Qb, float* __restrict__ Vb) {
  const int which = blockIdx.y;
  const float* W    = (which == 0) ? Wk : (which == 1) ? Wq : Wv;
  const float* bias = (which == 0) ? bk : (which == 1) ? bq : bv;
  float*       outp = (which == 0) ? Kb : (which == 1) ? Qb : Vb;
  const int row0 = blockIdx.x * 64;
  gemm64(x, W, row0, threadIdx.x,
         [&](int row, int col, float acc) {
           outp[(size_t)row * OUTD + col] = acc + bias[col];
         });
}

// Output projection + gated skip.  grid=ceil(N/64), block=256.
__global__ __launch_bounds__(256) void out_kernel(
    const float* __restrict__ tmean, const float* __restrict__ Wa,
    const float* __restrict__ ba, const float* __restrict__ x,
    const float* __restrict__ skip, float* __restrict__ out) {
  const int   row0  = blockIdx.x * 64;
  const float alpha = 1.0f / (1.0f + __expf(-skip[0]));
  const float beta  = 1.0f - alpha;
  gemm64(tmean, Wa, row0, threadIdx.x,
         [&](int row, int col, float acc) {
           const size_t idx = (size_t)row * OUTD + col;
           out[idx] = alpha * (acc + ba[col]) + beta * x[idx];
         });
}

// ---------------------------------------------------------------------------
// Edge attention.  grid=(E/16, R), block=32 (one wave per 16-edge tile).
// Per head: key_e tile = gather(K[src]) @ Watt[r,h] via 4 WMMAs, then
// att = <q[dst], key_e> reduced across half-wave lanes; atomicMax seg max.
__global__ __launch_bounds__(32) void edge_att_kernel(
    const float* __restrict__ Kb, const float* __restrict__ Qb,
    const float* __restrict__ Watt, const float* __restrict__ pri,
    const int* __restrict__ src, const int* __restrict__ dst,
    float* __restrict__ att, unsigned* __restrict__ mEnc,
    unsigned* __restrict__ cnt) {
  const int tile = blockIdx.x;
  const int r    = blockIdx.y;
  const int l    = threadIdx.x;
  const int m    = l & 15;
  const int hi   = l >> 4;
  const int e0   = tile * 16;
  const int myS  = src[(size_t)r * EEe + e0 + m];
  const int myD  = dst[(size_t)r * EEe + e0 + m];
  if (l < 16) atomicAdd(&cnt[(size_t)r * Nn + myD], 1u);

  const float* krow = Kb + (size_t)myS * OUTD;
  for (int h = 0; h < HH; ++h) {
    const float* Wah = Watt + (size_t)(r * HH + h) * DKK * DKK;
    v8f c = zero8();
#pragma unroll
    for (int k0 = 0; k0 < DKK; k0 += 4) {
      const int kk = k0 + 2 * hi;
      v2f a, b;
      a[0] = krow[h * DKK + kk];
      a[1] = krow[h * DKK + kk + 1];
      b[0] = Wah[kk * DKK + m];
      b[1] = Wah[(kk + 1) * DKK + m];
      c = wmma4(a, b, c);
    }
    const float scale = pri[r * HH + h] * INV_SQRT_DK;
#pragma unroll
    for (int v = 0; v < 8; ++v) {
      const int   em = v + 8 * hi;                     // edge within tile
      const int   dm = __shfl(myD, em, 32);            // dst node of that edge
      const float qe = Qb[(size_t)dm * OUTD + h * DKK + m];
      float p = c[v] * qe;                             // key_e[em][m] * q[dm][m]
      p += __shfl_xor(p, 1, 32);
      p += __shfl_xor(p, 2, 32);
      p += __shfl_xor(p, 4, 32);
      p += __shfl_xor(p, 8, 32);                       // sum over 16 lanes/half
      const float aval = p * scale;
      if (m == 0) {
        att[((size_t)r * EEe + e0 + em) * HH + h] = aval;
        atomicMax(&mEnc[((size_t)r * Nn + dm) * HH + h], encF(aval));
      }
    }
  }
}

// ---------------------------------------------------------------------------
// ex = exp(att - m[seg]); denom[seg] += ex.  1D over R*E*H.
__global__ __launch_bounds__(256) void softmax_kernel(
    const int* __restrict__ dst, const unsigned* __restrict__ mEnc,
    float* __restrict__ att, float* __restrict__ denom) {
  const size_t idx = (size_t)blockIdx.x * 256 + threadIdx.x;
  if (idx >= (size_t)RRr * EEe * HH) return;
  const int    h = (int)(idx & (HH - 1));
  const size_t t = idx >> 3;                 // r*E + e
  const int    r = (int)(t / EEe);
  const int    d = dst[t];
  const float  mm = decF(mEnc[((size_t)r * Nn + d) * HH + h]);
  const float  ex = __expf(att[idx] - mm);
  att[idx] = ex;
  atomicAdd(&denom[((size_t)r * Nn + d) * HH + h], ex);
}

// ---------------------------------------------------------------------------
// Aggregate: val_e tile = gather(V[src]) @ Wmsg[r,h] via WMMA; scatter
// tsum[dst] += (ex/denom) * val_e (sum over relations folded in directly).
__global__ __launch_bounds__(32) void edge_agg_kernel(
    const float* __restrict__ Vb, const float* __restrict__ Wmsg,
    const int* __restrict__ src, const int* __restrict__ dst,
    const float* __restrict__ att, const float* __restrict__ denom,
    float* __restrict__ tsum) {
  const int tile = blockIdx.x;
  const int r    = blockIdx.y;
  const int l    = threadIdx.x;
  const int m    = l & 15;
  const int hi   = l >> 4;
  const int e0   = tile * 16;
  const int myS  = src[(size_t)r * EEe + e0 + m];
  const int myD  = dst[(size_t)r * EEe + e0 + m];

  const float* vrow = Vb + (size_t)myS * OUTD;
  for (int h = 0; h < HH; ++h) {
    const float* Wmh = Wmsg + (size_t)(r * HH + h) * DKK * DKK;
    v8f c = zero8();
#pragma unroll
    for (int k0 = 0; k0 < DKK; k0 += 4) {
      const int kk = k0 + 2 * hi;
      v2f a, b;
      a[0] = vrow[h * DKK + kk];
      a[1] = vrow[h * DKK + kk + 1];
      b[0] = Wmh[kk * DKK + m];
      b[1] = Wmh[(kk + 1) * DKK + m];
      c = wmma4(a, b, c);
    }
#pragma unroll
    for (int v = 0; v < 8; ++v) {
      const int   em  = v + 8 * hi;
      const int   dm  = __shfl(myD, em, 32);
      const float ex  = att[((size_t)r * EEe + e0 + em) * HH + h];
      const float den = denom[((size_t)r * Nn + dm) * HH + h];
      const float w   = ex / den;
      atomicAdd(&tsum[(size_t)dm * OUTD + h * DKK + m], c[v] * w);
    }
  }
}

// ---------------------------------------------------------------------------
// t_mean = tsum / max(#relations with messages, 1)
__global__ __launch_bounds__(256) void finalize_kernel(
    const float* __restrict__ tsum, const unsigned* __restrict__ cnt,
    float* __restrict__ tmean) {
  const size_t idx = (size_t)blockIdx.x * 256 + threadIdx.x;
  if (idx >= (size_t)Nn * OUTD) return;
  const int n  = (int)(idx >> 7);
  int nz = (cnt[n] > 0) + (cnt[(size_t)Nn + n] > 0) +
           (cnt[(size_t)2 * Nn + n] > 0) + (cnt[(size_t)3 * Nn + n] > 0);
  if (nz < 1) nz = 1;
  tmean[idx] = tsum[idx] / (float)nz;
}

}  // namespace

// ---------------------------------------------------------------------------
extern "C" void kernel_launch(void* const* d_in, const int* in_sizes, int n_in,
                              void* d_out, int out_size, void* d_ws, size_t ws_size,
                              hipStream_t stream) {
  const float* x    = (const float*)d_in[0];
  const float* Wk   = (const float*)d_in[1];
  const float* bk   = (const float*)d_in[2];
  const float* Wq   = (const float*)d_in[3];
  const float* bq   = (const float*)d_in[4];
  const float* Wv   = (const float*)d_in[5];
  const float* bv   = (const float*)d_in[6];
  const float* Watt = (const float*)d_in[7];
  const float* Wmsg = (const float*)d_in[8];
  const float* pri  = (const float*)d_in[9];
  const float* Wa   = (const float*)d_in[10];
  const float* ba   = (const float*)d_in[11];
  const float* skip = (const float*)d_in[12];
  const int*   src  = (const int*)d_in[13];
  const int*   dst  = (const int*)d_in[14];
  float*       out  = (float*)d_out;

  // Workspace layout (fp32 words), ~250 MB total.
  float* ws = (float*)d_ws;
  size_t o = 0;
  float*    Kb   = ws + o; o += (size_t)Nn * OUTD;        // reused as t_mean
  float*    Qb   = ws + o; o += (size_t)Nn * OUTD;
  float*    Vb   = ws + o; o += (size_t)Nn * OUTD;
  float*    attB = ws + o; o += (size_t)RRr * EEe * HH;
  unsigned* mB   = (unsigned*)(ws + o); o += (size_t)RRr * Nn * HH;
  float*    denB = ws + o; o += (size_t)RRr * Nn * HH;
  float*    tsum = ws + o; o += (size_t)Nn * OUTD;
  unsigned* cntB = (unsigned*)(ws + o); o += (size_t)RRr * Nn;
  float*    tmean = Kb;  // Kb dead after edge_att; reuse for t_mean

  // Zero [mB .. cntB] (contiguous): m(=encoded -inf is 0), denom, tsum, cnt.
  const size_t zeroN = (size_t)RRr * Nn * HH * 2 + (size_t)Nn * OUTD +
                       (size_t)RRr * Nn;
  init_zero_kernel<<<4096, 256, 0, stream>>>(mB, zeroN);

  const unsigned nrt = (unsigned)((Nn + 63) / 64);  // 1563 row-tiles of 64
  kqv_kernel<<<dim3(nrt, 3), 256, 0, stream>>>(
      x, Wk, bk, Wq, bq, Wv, bv, Kb, Qb, Vb);

  edge_att_kernel<<<dim3(EEe / 16, RRr), 32, 0, stream>>>(
      Kb, Qb, Watt, pri, src, dst, attB, mB, cntB);

  softmax_kernel<<<(unsigned)(((size_t)RRr * EEe * HH + 255) / 256), 256, 0,
                   stream>>>(dst, mB, attB, denB);

  edge_agg_kernel<<<dim3(EEe / 16, RRr), 32, 0, stream>>>(
      Vb, Wmsg, src, dst, attB, denB, tsum);

  finalize_kernel<<<(unsigned)(((size_t)Nn * OUTD + 255) / 256), 256, 0,
                    stream>>>(tsum, cntB, tmean);

  out_kernel<<<nrt, 256, 0, stream>>>(tmean, Wa, ba, x, skip, out);
}